// WeightedBCELoss_76433237999935
// MI455X (gfx1250) — compile-verified
//
#include <hip/hip_runtime.h>

// ---- types --------------------------------------------------------------
typedef __attribute__((ext_vector_type(2))) float v2f;
typedef __attribute__((ext_vector_type(8))) float v8f;
typedef __attribute__((ext_vector_type(4))) float vf4;
typedef __attribute__((ext_vector_type(4))) int   vi4;

#define POS_W 1.0f
#define NEG_W 8.0f
#define EPS_F 1e-7f

#define BLOCKS 2048
#define TPB    256
#define WAVES_PER_BLOCK (TPB / 32)

// ---- kernel 1: streaming elementwise + per-block partial sum ------------
// Memory-bound: 8 B/element, non-temporal b128 loads (stream > 192MB L2).
__global__ void bce_partial_kernel(const float* __restrict__ probs,
                                   const int*   __restrict__ tgts,
                                   float* __restrict__ partials,
                                   int n) {
    __shared__ float s_wave[WAVES_PER_BLOCK];

    const vf4* __restrict__ p4 = reinterpret_cast<const vf4*>(probs);
    const vi4* __restrict__ t4 = reinterpret_cast<const vi4*>(tgts);

    const int n4     = n >> 2;
    const int stride = gridDim.x * blockDim.x;
    float acc = 0.0f;

    for (int i = blockIdx.x * blockDim.x + threadIdx.x; i < n4; i += stride) {
        vf4 x = __builtin_nontemporal_load(&p4[i]);   // global_load_b128 th:NT
        vi4 t = __builtin_nontemporal_load(&t4[i]);
#pragma unroll
        for (int j = 0; j < 4; ++j) {
            const float xv  = x[j];
            const bool  pos = (t[j] == 1);
            // select log argument and weight BEFORE the log: one v_log per elem
            const float arg = pos ? (xv + EPS_F) : (1.0f - xv + EPS_F);
            const float w   = pos ? POS_W : NEG_W;
            acc = __builtin_fmaf(w, -__logf(arg), acc);
        }
    }

    // wave32 butterfly reduction (fixed order -> deterministic)
#pragma unroll
    for (int off = 16; off > 0; off >>= 1)
        acc += __shfl_xor(acc, off, 32);

    const int lane = threadIdx.x & 31;
    const int wave = threadIdx.x >> 5;
    if (lane == 0) s_wave[wave] = acc;        // ds_store_b32
    __syncthreads();

    if (threadIdx.x == 0) {
        float bsum = 0.0f;
#pragma unroll
        for (int w = 0; w < WAVES_PER_BLOCK; ++w)
            bsum += s_wave[w];                 // ds_load_b32, fixed order
        partials[blockIdx.x] = bsum;
    }
}

// ---- kernel 2: single-wave final reduction using V_WMMA_F32_16X16X4_F32 -
// A (16x4 f32) holds one partial per lane in VGPR0 (VGPR1 = 0); B = ones.
// D[m][n] = A-row-m sum = s[lane m] + s[lane m+16]  (exact f32 products).
// Lane L's 8 D registers are rows L<16 ? 0..7 : 8..15, each replicated
// across columns; t(lane0)+t(lane16) = total of all 32 lane partials.
__global__ void bce_final_kernel(const float* __restrict__ partials,
                                 float* __restrict__ out,
                                 int n, int nPartials) {
    const int lane = threadIdx.x;              // exactly 32 threads: EXEC all 1s

    float s = 0.0f;
    const int per_lane = nPartials >> 5;       // 2048/32 = 64, fixed order
    for (int i = 0; i < per_lane; ++i)
        s += partials[lane + (i << 5)];

    v2f a; a.x = s;    a.y = 0.0f;             // A: 16x4, K=0 holds partials
    v2f b; b.x = 1.0f; b.y = 1.0f;             // B: 4x16, all ones
    v8f c = {};
    v8f d = __builtin_amdgcn_wmma_f32_16x16x4_f32(
        /*neg_a=*/false, a, /*neg_b=*/false, b,
        /*c_mod=*/(short)0, c, /*reuse_a=*/false, /*reuse_b=*/false);

    float t = 0.0f;
#pragma unroll
    for (int r = 0; r < 8; ++r) t += d[r];     // rows 0-7 (lanes<16) / 8-15

    const float t16  = __shfl(t, 16, 32);      // lane0 pulls rows 8-15 total
    if (lane == 0)
        out[0] = (t + t16) / (float)n;         // mean
}

// ---- launch -------------------------------------------------------------
extern "C" void kernel_launch(void* const* d_in, const int* in_sizes, int n_in,
                              void* d_out, int out_size, void* d_ws, size_t ws_size,
                              hipStream_t stream) {
    const float* probs = (const float*)d_in[0];
    const int*   tgts  = (const int*)d_in[1];
    float*       outp  = (float*)d_out;
    float*       part  = (float*)d_ws;         // BLOCKS floats = 8 KB scratch
    const int n = in_sizes[0];                 // 33554432, divisible by 4

    bce_partial_kernel<<<BLOCKS, TPB, 0, stream>>>(probs, tgts, part, n);
    bce_final_kernel<<<1, 32, 0, stream>>>(part, outp, n, BLOCKS);
}